// PartialEncoderWeightedSumEDDIMultiWeightATSE_57767310131608
// MI455X (gfx1250) — compile-verified
//
#include <hip/hip_runtime.h>

// Problem constants (from reference)
#define Bn 64
#define Jn 20000
#define JT 1250      // 16-row j tiles
#define SLOTS 64     // deterministic partial-reduction slots per batch row

typedef float v2f __attribute__((ext_vector_type(2)));
typedef float v8f __attribute__((ext_vector_type(8)));

// V_WMMA_F32_16X16X4_F32 : D = A(16x4 f32) x B(4x16 f32) + C(16x16 f32)
__device__ __forceinline__ v8f wmma_f32(v2f a, v2f b, v8f c) {
  return __builtin_amdgcn_wmma_f32_16x16x4_f32(false, a, false, b, (short)0, c,
                                               false, false);
}

__device__ __forceinline__ float wred32(float v) {
#pragma unroll
  for (int o = 16; o > 0; o >>= 1) v += __shfl_xor(v, o, 32);
  return v;
}

// ---------------------------------------------------------------------------
// Phase 0: P[j,64] = fe[j] @ h_W1[1:33,:]   (j-only part of h layer 1)
//          Q[j,24] = ae[idx[j]] @ g_W1[32:48,:] (j-only part of gate layer 1)
// 4 waves per block, one 16-row j tile per wave.
// ---------------------------------------------------------------------------
__global__ __launch_bounds__(128) void k_pre(
    const float* __restrict__ fe, const float* __restrict__ ae,
    const int* __restrict__ aidx, const float* __restrict__ hW1,
    const float* __restrict__ gW1, float* __restrict__ Pg,
    float* __restrict__ Qg) {
  __shared__ float W1s[32][72];   // h_W1 rows 1..32, stride 72 (2s%64==16 -> no conflicts)
  __shared__ float GW1b[16][40];  // g_W1 rows 32..47, cols padded to 40 with 0
  __shared__ float fes[4][16][36];
  __shared__ float aes[4][16][20];
  __shared__ int idxs[4][16];

  const int tid = threadIdx.x, lane = tid & 31, wid = tid >> 5;
  for (int i = tid; i < 32 * 72; i += 128) {
    int k = i / 72, n = i % 72;
    W1s[k][n] = (n < 64) ? hW1[(1 + k) * 64 + n] : 0.f;
  }
  for (int i = tid; i < 16 * 40; i += 128) {
    int k = i / 40, n = i % 40;
    GW1b[k][n] = (n < 24) ? gW1[(32 + k) * 24 + n] : 0.f;
  }
  __syncthreads();

  const int tile = blockIdx.x * 4 + wid;
  if (tile >= JT) return;
  const int j0 = tile * 16;
  const int row16 = lane & 15, hl = lane >> 4, khalf = hl * 2, col16 = lane & 15;

  // stage fe rows (contiguous 512 floats) and gathered ae rows
#pragma unroll
  for (int t = 0; t < 16; t++) {
    int i = t * 32 + lane;
    fes[wid][i >> 5][i & 31] = fe[(size_t)j0 * 32 + i];
  }
  if (lane < 16) idxs[wid][lane] = aidx[j0 + lane];
  __builtin_amdgcn_wave_barrier();
#pragma unroll
  for (int t = 0; t < 8; t++) {
    int i = t * 32 + lane, r = i >> 4, c = i & 15;
    aes[wid][r][c] = ae[(size_t)idxs[wid][r] * 16 + c];
  }
  __builtin_amdgcn_wave_barrier();

  // P tile: [16x32] @ [32x64]  -> 4 N-tiles x 8 K-steps
  v8f cP[4];
#pragma unroll
  for (int nt = 0; nt < 4; nt++) cP[nt] = (v8f){0.f, 0.f, 0.f, 0.f, 0.f, 0.f, 0.f, 0.f};
#pragma unroll
  for (int kk = 0; kk < 8; kk++) {
    const int kb = kk * 4 + khalf;
    v2f a;
    a.x = fes[wid][row16][kb];
    a.y = fes[wid][row16][kb + 1];
#pragma unroll
    for (int nt = 0; nt < 4; nt++) {
      v2f bf;
      bf.x = W1s[kb][nt * 16 + col16];
      bf.y = W1s[kb + 1][nt * 16 + col16];
      cP[nt] = wmma_f32(a, bf, cP[nt]);
    }
  }
#pragma unroll
  for (int nt = 0; nt < 4; nt++)
#pragma unroll
    for (int v = 0; v < 8; v++)
      Pg[(size_t)(j0 + v + 8 * hl) * 64 + nt * 16 + col16] = cP[nt][v];

  // Q tile: [16x16] @ [16x24] -> 2 N-tiles x 4 K-steps
  v8f cQ[2];
  cQ[0] = (v8f){0.f, 0.f, 0.f, 0.f, 0.f, 0.f, 0.f, 0.f};
  cQ[1] = cQ[0];
#pragma unroll
  for (int kk = 0; kk < 4; kk++) {
    const int kb = kk * 4 + khalf;
    v2f a;
    a.x = aes[wid][row16][kb];
    a.y = aes[wid][row16][kb + 1];
#pragma unroll
    for (int nt = 0; nt < 2; nt++) {
      v2f bf;
      bf.x = GW1b[kb][nt * 16 + col16];
      bf.y = GW1b[kb + 1][nt * 16 + col16];
      cQ[nt] = wmma_f32(a, bf, cQ[nt]);
    }
  }
#pragma unroll
  for (int nt = 0; nt < 2; nt++)
#pragma unroll
    for (int v = 0; v < 8; v++) {
      int col = nt * 16 + col16;
      if (col < 24) Qg[(size_t)(j0 + v + 8 * hl) * 24 + col] = cQ[nt][v];
    }
}

// ---------------------------------------------------------------------------
// Phase 1: per (b, j-tile): LN1 -> h2 (WMMA) -> LN2 -> gate1 (WMMA) ->
// gate2 (WMMA) -> e=mask*exp(clip(logit)); single-pass softmax accumulation
// into deterministic per-slot partials.
// ---------------------------------------------------------------------------
__global__ __launch_bounds__(128) void k_main(
    const float* __restrict__ x, const int* __restrict__ mask,
    const float* __restrict__ Pg, const float* __restrict__ Qg,
    const float* __restrict__ hW1, const float* __restrict__ hb1,
    const float* __restrict__ l1g, const float* __restrict__ l1b,
    const float* __restrict__ hW2, const float* __restrict__ hb2,
    const float* __restrict__ l2g, const float* __restrict__ l2b,
    const float* __restrict__ gW1, const float* __restrict__ gb1,
    const float* __restrict__ gW2, const float* __restrict__ gb2,
    float* __restrict__ partial) {
  __shared__ float w0s[64], hb1s[64], l1gs[64], l1bs[64];
  __shared__ float W2s[64][40];  // h_W2 [64x32] padded; 2*40%64==16 -> conflict-free B frags
  __shared__ float hb2s[32], l2gs[32], l2bs[32];
  __shared__ float GW1a[32][40];  // g_W1 rows 0..31 (h_out part), padded
  __shared__ float gb1s[24];
  __shared__ float GW2s[24][16];  // g_W2 [24x4] padded to 16 cols
  __shared__ float gb2s[4];
  __shared__ float A1s[4][16][68];  // h1 activations per wave
  __shared__ float A2s[4][16][36];  // h2 pre-LN / activations per wave
  __shared__ float G1s[4][16][28];  // gate1 activations per wave
  __shared__ float QsS[4][384];     // staged Q rows
  __shared__ __align__(16) float Es[4][16][4];
  __shared__ float Xs[4][16];
  __shared__ int Msk[4][16];

  const int tid = threadIdx.x, lane = tid & 31, wid = tid >> 5;
  const int b = blockIdx.y;

  for (int i = tid; i < 64; i += 128) {
    w0s[i] = hW1[i];  // h_W1 row 0 (coefficient of x)
    hb1s[i] = hb1[i];
    l1gs[i] = l1g[i];
    l1bs[i] = l1b[i];
  }
  for (int i = tid; i < 64 * 40; i += 128) {
    int k = i / 40, n = i % 40;
    W2s[k][n] = (n < 32) ? hW2[k * 32 + n] : 0.f;
  }
  for (int i = tid; i < 32; i += 128) {
    hb2s[i] = hb2[i];
    l2gs[i] = l2g[i];
    l2bs[i] = l2b[i];
  }
  for (int i = tid; i < 32 * 40; i += 128) {
    int k = i / 40, n = i % 40;
    GW1a[k][n] = (n < 24) ? gW1[k * 24 + n] : 0.f;
  }
  for (int i = tid; i < 24; i += 128) gb1s[i] = gb1[i];
  for (int i = tid; i < 24 * 16; i += 128) {
    int k = i >> 4, n = i & 15;
    GW2s[k][n] = (n < 4) ? gW2[k * 4 + n] : 0.f;
  }
  if (tid < 4) gb2s[tid] = gb2[tid];
  __syncthreads();

  const int row16 = lane & 15, hl = lane >> 4, khalf = hl * 2, col16 = lane & 15;
  float num0 = 0.f, num1 = 0.f, num2 = 0.f, num3 = 0.f;
  float den0 = 0.f, den1 = 0.f, den2 = 0.f, den3 = 0.f;
  const int slot = blockIdx.x * 4 + wid;  // 0..63

  for (int tile = slot; tile < JT; tile += SLOTS) {
    const int j0 = tile * 16;
    if (lane < 16) {
      Xs[wid][lane] = x[(size_t)b * Jn + j0 + lane];
      Msk[wid][lane] = mask[(size_t)b * Jn + j0 + lane];
    }
#pragma unroll
    for (int t = 0; t < 12; t++) {
      int i = t * 32 + lane;
      QsS[wid][i] = Qg[(size_t)j0 * 24 + i];
    }
    __builtin_amdgcn_wave_barrier();
    if (tile + SLOTS < JT)
      __builtin_prefetch(Pg + (size_t)(tile + SLOTS) * 16 * 64, 0, 1);

    // ---- layer 1: v = x*w0 + P + b1; LN over 64 cols; relu -> A1s ----
    {
      const float4* P4 =
          (const float4*)(Pg + (size_t)(j0 + row16) * 64 + hl * 32);
      const float xr = Xs[wid][row16];
      float vbuf[32];
      float s = 0.f, q = 0.f;
#pragma unroll
      for (int t = 0; t < 8; t++) {
        float4 p = P4[t];
        int c = hl * 32 + t * 4;
        float a0 = fmaf(xr, w0s[c + 0], p.x + hb1s[c + 0]);
        float a1 = fmaf(xr, w0s[c + 1], p.y + hb1s[c + 1]);
        float a2 = fmaf(xr, w0s[c + 2], p.z + hb1s[c + 2]);
        float a3 = fmaf(xr, w0s[c + 3], p.w + hb1s[c + 3]);
        vbuf[t * 4 + 0] = a0;
        vbuf[t * 4 + 1] = a1;
        vbuf[t * 4 + 2] = a2;
        vbuf[t * 4 + 3] = a3;
        s += a0 + a1 + a2 + a3;
        q += a0 * a0 + a1 * a1 + a2 * a2 + a3 * a3;
      }
      s += __shfl_xor(s, 16, 32);
      q += __shfl_xor(q, 16, 32);
      const float mean = s * (1.f / 64.f);
      const float var = q * (1.f / 64.f) - mean * mean;
      const float rstd = rsqrtf(var + 1e-5f);
#pragma unroll
      for (int i = 0; i < 32; i++) {
        int c = hl * 32 + i;
        float a = fmaf((vbuf[i] - mean) * rstd, l1gs[c], l1bs[c]);
        A1s[wid][row16][c] = fmaxf(a, 0.f);
      }
    }
    __builtin_amdgcn_wave_barrier();

    // ---- layer 2: [16x64] @ [64x32] via WMMA ----
    v8f c2[2];
    c2[0] = (v8f){0.f, 0.f, 0.f, 0.f, 0.f, 0.f, 0.f, 0.f};
    c2[1] = c2[0];
#pragma unroll
    for (int kk = 0; kk < 16; kk++) {
      const int kb = kk * 4 + khalf;
      v2f a;
      a.x = A1s[wid][row16][kb];
      a.y = A1s[wid][row16][kb + 1];
#pragma unroll
      for (int nt = 0; nt < 2; nt++) {
        v2f bf;
        bf.x = W2s[kb][nt * 16 + col16];
        bf.y = W2s[kb + 1][nt * 16 + col16];
        c2[nt] = wmma_f32(a, bf, c2[nt]);
      }
    }
#pragma unroll
    for (int nt = 0; nt < 2; nt++)
#pragma unroll
      for (int v = 0; v < 8; v++)
        A2s[wid][v + 8 * hl][nt * 16 + col16] = c2[nt][v];
    __builtin_amdgcn_wave_barrier();

    // ---- LN2 over 32 cols + relu (in place) ----
    {
      float t2[16];
      float s = 0.f, q = 0.f;
#pragma unroll
      for (int i = 0; i < 16; i++) {
        int c = hl * 16 + i;
        float hv = A2s[wid][row16][c] + hb2s[c];
        t2[i] = hv;
        s += hv;
        q += hv * hv;
      }
      s += __shfl_xor(s, 16, 32);
      q += __shfl_xor(q, 16, 32);
      const float mean = s * (1.f / 32.f);
      const float var = q * (1.f / 32.f) - mean * mean;
      const float rstd = rsqrtf(var + 1e-5f);
#pragma unroll
      for (int i = 0; i < 16; i++) {
        int c = hl * 16 + i;
        float a = fmaf((t2[i] - mean) * rstd, l2gs[c], l2bs[c]);
        A2s[wid][row16][c] = fmaxf(a, 0.f);
      }
    }
    __builtin_amdgcn_wave_barrier();

    // ---- gate layer 1: [16x32] @ [32x24] + Q + b, relu ----
    v8f cg1[2];
    cg1[0] = (v8f){0.f, 0.f, 0.f, 0.f, 0.f, 0.f, 0.f, 0.f};
    cg1[1] = cg1[0];
#pragma unroll
    for (int kk = 0; kk < 8; kk++) {
      const int kb = kk * 4 + khalf;
      v2f a;
      a.x = A2s[wid][row16][kb];
      a.y = A2s[wid][row16][kb + 1];
#pragma unroll
      for (int nt = 0; nt < 2; nt++) {
        v2f bf;
        bf.x = GW1a[kb][nt * 16 + col16];
        bf.y = GW1a[kb + 1][nt * 16 + col16];
        cg1[nt] = wmma_f32(a, bf, cg1[nt]);
      }
    }
#pragma unroll
    for (int nt = 0; nt < 2; nt++)
#pragma unroll
      for (int v = 0; v < 8; v++) {
        int r = v + 8 * hl;
        int col = nt * 16 + col16;
        if (col < 24) {
          float val = cg1[nt][v] + QsS[wid][r * 24 + col] + gb1s[col];
          G1s[wid][r][col] = fmaxf(val, 0.f);
        }
      }
    __builtin_amdgcn_wave_barrier();

    // ---- gate layer 2: [16x24] @ [24x4] ----
    v8f c3 = (v8f){0.f, 0.f, 0.f, 0.f, 0.f, 0.f, 0.f, 0.f};
#pragma unroll
    for (int kk = 0; kk < 6; kk++) {
      const int kb = kk * 4 + khalf;
      v2f a;
      a.x = G1s[wid][row16][kb];
      a.y = G1s[wid][row16][kb + 1];
      v2f bf;
      bf.x = GW2s[kb][col16];
      bf.y = GW2s[kb + 1][col16];
      c3 = wmma_f32(a, bf, c3);
    }
    // e = mask * exp(clip(logit, -10, 10))  (clip makes max-free softmax safe)
#pragma unroll
    for (int v = 0; v < 8; v++) {
      int r = v + 8 * hl;
      if (col16 < 4) {
        float lg = c3[v] + gb2s[col16];
        lg = fminf(10.f, fmaxf(-10.f, lg));
        Es[wid][r][col16] = Msk[wid][r] ? __expf(lg) : 0.f;
      }
    }
    __builtin_amdgcn_wave_barrier();

    // ---- single-pass weighted accumulation: lane owns column d=lane ----
#pragma unroll
    for (int r = 0; r < 16; r++) {
      float a = A2s[wid][r][lane];
      float4 e4 = *(const float4*)&Es[wid][r][0];
      num0 = fmaf(e4.x, a, num0);
      num1 = fmaf(e4.y, a, num1);
      num2 = fmaf(e4.z, a, num2);
      num3 = fmaf(e4.w, a, num3);
      den0 += e4.x;
      den1 += e4.y;
      den2 += e4.z;
      den3 += e4.w;
    }
    __builtin_amdgcn_wave_barrier();
  }

  // deterministic per-slot partials: [b][slot][132] = {num[4][32], den[4]}
  float* pp = partial + ((size_t)b * SLOTS + slot) * 132;
  pp[0 * 32 + lane] = num0;
  pp[1 * 32 + lane] = num1;
  pp[2 * 32 + lane] = num2;
  pp[3 * 32 + lane] = num3;
  float dsel = (lane == 0) ? den0 : ((lane == 1) ? den1 : ((lane == 2) ? den2 : den3));
  if (lane < 4) pp[128 + lane] = dsel;
}

// ---------------------------------------------------------------------------
// Phase 2: reduce slots; combiner + encoder MLP tail (64 rows, 1 wave each).
// ---------------------------------------------------------------------------
__global__ __launch_bounds__(32) void k_tail(
    const float* __restrict__ partial, const float* __restrict__ cW,
    const float* __restrict__ cb, const float* __restrict__ cg,
    const float* __restrict__ cbb, const float* __restrict__ eW1,
    const float* __restrict__ eb1, const float* __restrict__ eW2,
    const float* __restrict__ eb2, float* __restrict__ out) {
  const int b = blockIdx.x, l = threadIdx.x;
  __shared__ float hs[128], comb[32], e1s[128], denS[4];

  float numsum[4] = {0.f, 0.f, 0.f, 0.f};
  float densum = 0.f;
  for (int slot = 0; slot < SLOTS; slot++) {
    const float* pp = partial + ((size_t)b * SLOTS + slot) * 132;
#pragma unroll
    for (int t = 0; t < 4; t++) numsum[t] += pp[t * 32 + l];
    if (l < 4) densum += pp[128 + l];
  }
  if (l < 4) denS[l] = densum;
  __syncthreads();
  const bool any = denS[0] > 0.f;  // mask identical across heads
#pragma unroll
  for (int t = 0; t < 4; t++)
    hs[t * 32 + l] = (denS[t] > 0.f) ? (numsum[t] / denS[t]) : 0.f;
  __syncthreads();

  // combined = relu(LN(hs @ c_W + c_b) * g + b), zeroed if no observed j
  float s = cb[l];
  for (int i = 0; i < 128; i++) s = fmaf(hs[i], cW[i * 32 + l], s);
  float m = wred32(s) * (1.f / 32.f);
  float vq = wred32(s * s) * (1.f / 32.f) - m * m;
  float rstd = rsqrtf(vq + 1e-5f);
  float cv = fmaxf(fmaf((s - m) * rstd, cg[l], cbb[l]), 0.f);
  if (!any) cv = 0.f;
  comb[l] = cv;
  __syncthreads();

  // e1 = relu(LN(combined @ e_W1 + e_b1))   (LN no affine)
  float s1[4];
  float ps = 0.f, pq = 0.f;
#pragma unroll
  for (int t = 0; t < 4; t++) {
    int c = t * 32 + l;
    float v = eb1[c];
    for (int i = 0; i < 32; i++) v = fmaf(comb[i], eW1[i * 128 + c], v);
    s1[t] = v;
    ps += v;
    pq += v * v;
  }
  m = wred32(ps) * (1.f / 128.f);
  vq = wred32(pq) * (1.f / 128.f) - m * m;
  rstd = rsqrtf(vq + 1e-5f);
#pragma unroll
  for (int t = 0; t < 4; t++) e1s[t * 32 + l] = fmaxf((s1[t] - m) * rstd, 0.f);
  __syncthreads();

  // mu_logvar = relu(LN(e1 @ e_W2 + e_b2)); split -> mu, logvar
  float s2[2];
  ps = 0.f;
  pq = 0.f;
#pragma unroll
  for (int t = 0; t < 2; t++) {
    int c = t * 32 + l;
    float v = eb2[c];
    for (int i = 0; i < 128; i++) v = fmaf(e1s[i], eW2[i * 64 + c], v);
    s2[t] = v;
    ps += v;
    pq += v * v;
  }
  m = wred32(ps) * (1.f / 64.f);
  vq = wred32(pq) * (1.f / 64.f) - m * m;
  rstd = rsqrtf(vq + 1e-5f);
  out[(size_t)b * 32 + l] = fmaxf((s2[0] - m) * rstd, 0.f);                // mu
  out[(size_t)Bn * 32 + (size_t)b * 32 + l] = fmaxf((s2[1] - m) * rstd, 0.f);  // logvar
}

// ---------------------------------------------------------------------------
extern "C" void kernel_launch(void* const* d_in, const int* in_sizes, int n_in,
                              void* d_out, int out_size, void* d_ws,
                              size_t ws_size, hipStream_t stream) {
  (void)in_sizes;
  (void)n_in;
  (void)out_size;
  (void)ws_size;
  const float* x = (const float*)d_in[0];
  const int* mask = (const int*)d_in[1];
  const int* aidx = (const int*)d_in[2];
  const float* fe = (const float*)d_in[3];
  const float* ae = (const float*)d_in[4];
  const float* hW1 = (const float*)d_in[5];
  const float* hb1 = (const float*)d_in[6];
  const float* l1g = (const float*)d_in[7];
  const float* l1b = (const float*)d_in[8];
  const float* hW2 = (const float*)d_in[9];
  const float* hb2 = (const float*)d_in[10];
  const float* l2g = (const float*)d_in[11];
  const float* l2b = (const float*)d_in[12];
  const float* gW1 = (const float*)d_in[13];
  const float* gb1 = (const float*)d_in[14];
  const float* gW2 = (const float*)d_in[15];
  const float* gb2 = (const float*)d_in[16];
  const float* cW = (const float*)d_in[17];
  const float* cb = (const float*)d_in[18];
  const float* cg = (const float*)d_in[19];
  const float* cbb = (const float*)d_in[20];
  const float* eW1 = (const float*)d_in[21];
  const float* eb1 = (const float*)d_in[22];
  const float* eW2 = (const float*)d_in[23];
  const float* eb2 = (const float*)d_in[24];

  float* Pg = (float*)d_ws;                 // [J][64]  = 1,280,000 f
  float* Qg = Pg + (size_t)Jn * 64;         // [J][24]  =   480,000 f
  float* partial = Qg + (size_t)Jn * 24;    // [B][64][132] = 540,672 f (~9.2 MB total)

  k_pre<<<dim3((JT + 3) / 4), 128, 0, stream>>>(fe, ae, aidx, hW1, gW1, Pg, Qg);
  k_main<<<dim3(SLOTS / 4, Bn), 128, 0, stream>>>(
      x, mask, Pg, Qg, hW1, hb1, l1g, l1b, hW2, hb2, l2g, l2b, gW1, gb1, gW2,
      gb2, partial);
  k_tail<<<dim3(Bn), 32, 0, stream>>>(partial, cW, cb, cg, cbb, eW1, eb1, eW2,
                                      eb2, (float*)d_out);
}